// MPNNAttn_68513318306405
// MI455X (gfx1250) — compile-verified
//
#include <hip/hip_runtime.h>
#include <math.h>

#define B_ 8
#define N_ 2048
#define F_ 73
#define H_ 128
#define M_ (B_*N_)   // 16384 rows
#define LT_ 12

typedef float v2f __attribute__((ext_vector_type(2)));
typedef float v8f __attribute__((ext_vector_type(8)));

#define WMMA_F32(a, b, c) __builtin_amdgcn_wmma_f32_16x16x4_f32( \
    false, (a), false, (b), (short)0, (c), false, false)

// ---------------- setup: pad h_in -> h0 (128 feat), copy to ht, node mask ----
__global__ void k_setup(const float* __restrict__ h_in, float* __restrict__ h0,
                        float* __restrict__ ht, float* __restrict__ mask) {
  int node = blockIdx.x;
  int t = threadIdx.x;
  float v = (t < F_) ? h_in[(size_t)node*F_ + t] : 0.f;
  h0[(size_t)node*H_ + t] = v;
  ht[(size_t)node*H_ + t] = v;
  __shared__ float sh[H_];
  sh[t] = v;
  __syncthreads();
  for (int s = H_/2; s > 0; s >>= 1) {
    if (t < s) sh[t] += sh[t+s];
    __syncthreads();
  }
  if (t == 0) mask[node] = (sh[0] > 0.f) ? 1.f : 0.f;
}

// ---------------- fp32 WMMA GEMM, 16x64 strip per wave ------------------------
// C[M,Nout] = X[M,K] @ W[Nout,K]^T (+bias)(+relu). K stepped by 16:
// all 20 operand loads (4 A frags + 16 B frags) are issued as a batch, then
// 16 chained WMMAs consume them -> clause-able loads, graduated s_wait_loadcnt.
// Grid sized exactly (no partial waves) so EXEC is all-ones at every WMMA.
__global__ void k_gemm_wmma4(const float* __restrict__ X, const float* __restrict__ W,
                             const float* __restrict__ bias, float* __restrict__ C,
                             int K, int Nout, int relu) {
  int wave = (int)((blockIdx.x * blockDim.x + threadIdx.x) >> 5);
  int lane = threadIdx.x & 31;
  int nstrips = Nout >> 6;                 // 64 cols per wave
  int m0 = (wave / nstrips) << 4;
  int n0 = (wave % nstrips) << 6;
  int ml = lane & 15;                      // A: row (M), B: col (N)
  int kk = (lane >> 4) << 1;               // lanes 0-15 -> K 0,1 ; 16-31 -> K 2,3
  const float* xp  = X + (size_t)(m0 + ml)*K + kk;
  const float* wp0 = W + (size_t)(n0      + ml)*K + kk;
  const float* wp1 = W + (size_t)(n0 + 16 + ml)*K + kk;
  const float* wp2 = W + (size_t)(n0 + 32 + ml)*K + kk;
  const float* wp3 = W + (size_t)(n0 + 48 + ml)*K + kk;
  __builtin_prefetch(xp, 0, 3);            // global_prefetch_b8
  __builtin_prefetch(wp0, 0, 3);
  __builtin_prefetch(wp1, 0, 3);
  __builtin_prefetch(wp2, 0, 3);
  __builtin_prefetch(wp3, 0, 3);

  v8f acc0 = (v8f){0.f,0.f,0.f,0.f,0.f,0.f,0.f,0.f};
  v8f acc1 = acc0, acc2 = acc0, acc3 = acc0;

  for (int k = 0; k < K; k += 16) {
    v2f a[4], b0[4], b1[4], b2[4], b3[4];
    #pragma unroll
    for (int u = 0; u < 4; ++u) {          // batch of 20 independent b64 loads
      a[u]  = *(const v2f*)(xp  + k + 4*u);
      b0[u] = *(const v2f*)(wp0 + k + 4*u);
      b1[u] = *(const v2f*)(wp1 + k + 4*u);
      b2[u] = *(const v2f*)(wp2 + k + 4*u);
      b3[u] = *(const v2f*)(wp3 + k + 4*u);
    }
    #pragma unroll
    for (int u = 0; u < 4; ++u) {          // 16 WMMAs consume the batch
      acc0 = WMMA_F32(a[u], b0[u], acc0);
      acc1 = WMMA_F32(a[u], b1[u], acc1);
      acc2 = WMMA_F32(a[u], b2[u], acc2);
      acc3 = WMMA_F32(a[u], b3[u], acc3);
    }
  }

  int mbase = m0 + ((lane >> 4) << 3);     // C/D: VGPR i -> row i (+8 hi lanes)
  float bv0 = bias ? bias[n0      + ml] : 0.f;
  float bv1 = bias ? bias[n0 + 16 + ml] : 0.f;
  float bv2 = bias ? bias[n0 + 32 + ml] : 0.f;
  float bv3 = bias ? bias[n0 + 48 + ml] : 0.f;
  #pragma unroll
  for (int i = 0; i < 8; ++i) {
    float v0 = acc0[i] + bv0, v1 = acc1[i] + bv1;
    float v2 = acc2[i] + bv2, v3 = acc3[i] + bv3;
    if (relu) { v0 = fmaxf(v0,0.f); v1 = fmaxf(v1,0.f); v2 = fmaxf(v2,0.f); v3 = fmaxf(v3,0.f); }
    size_t r = (size_t)(mbase + i)*Nout;
    C[r + n0      + ml] = v0;
    C[r + n0 + 16 + ml] = v1;
    C[r + n0 + 32 + ml] = v2;
    C[r + n0 + 48 + ml] = v3;
  }
}

// ---------------- fp32 WMMA GEMM, single 16x16 tile (Nout padded to 16) -------
__global__ void k_gemm_wmma1(const float* __restrict__ X, const float* __restrict__ W,
                             const float* __restrict__ bias, float* __restrict__ C,
                             int K, int NoutValid, int relu) {
  int wave = (int)((blockIdx.x * blockDim.x + threadIdx.x) >> 5);
  int lane = threadIdx.x & 31;
  int m0 = wave << 4;                      // one n-strip only (Nout<=16)
  int ml = lane & 15;
  int kk = (lane >> 4) << 1;
  const float* xp = X + (size_t)(m0 + ml)*K + kk;
  int wr = ml; if (wr > NoutValid - 1) wr = NoutValid - 1;   // clamp padded rows
  const float* wp = W + (size_t)wr*K + kk;
  __builtin_prefetch(xp, 0, 3);
  __builtin_prefetch(wp, 0, 3);
  v8f acc = (v8f){0.f,0.f,0.f,0.f,0.f,0.f,0.f,0.f};
  for (int k = 0; k < K; k += 16) {
    v2f a[4], b[4];
    #pragma unroll
    for (int u = 0; u < 4; ++u) {
      a[u] = *(const v2f*)(xp + k + 4*u);
      b[u] = *(const v2f*)(wp + k + 4*u);
    }
    #pragma unroll
    for (int u = 0; u < 4; ++u) acc = WMMA_F32(a[u], b[u], acc);
  }
  int n = ml;
  if (n < NoutValid) {
    float bv = bias ? bias[n] : 0.f;
    int mbase = m0 + ((lane >> 4) << 3);
    #pragma unroll
    for (int i = 0; i < 8; ++i) {
      float v = acc[i] + bv;
      if (relu) v = fmaxf(v, 0.f);
      C[(size_t)(mbase + i)*NoutValid + n] = v;
    }
  }
}

// ---------------- s1 = z@a1, s2 = z@a2, plus per-batch maxes ------------------
__global__ void k_attn_scores(const float* __restrict__ z, const float* __restrict__ attn_w,
                              float* __restrict__ s1, float* __restrict__ s2,
                              float* __restrict__ mx1, float* __restrict__ mx2) {
  int b = blockIdx.x;
  int t = threadIdx.x;
  __shared__ float sa1[H_], sa2[H_];
  if (t < H_) { sa1[t] = attn_w[t]; sa2[t] = attn_w[H_ + t]; }
  __syncthreads();
  const float* zb = z + (size_t)b*N_*H_;
  float l1 = -3.0e38f, l2 = -3.0e38f;
  for (int m = t; m < N_; m += 256) {
    const float* zr = zb + (size_t)m*H_;
    float d1 = 0.f, d2 = 0.f;
    #pragma unroll 8
    for (int k = 0; k < H_; ++k) { float zv = zr[k]; d1 += zv*sa1[k]; d2 += zv*sa2[k]; }
    s1[b*N_+m] = d1; s2[b*N_+m] = d2;
    l1 = fmaxf(l1, d1); l2 = fmaxf(l2, d2);
  }
  __shared__ float r1[256], r2[256];
  r1[t] = l1; r2[t] = l2; __syncthreads();
  for (int s = 128; s > 0; s >>= 1) {
    if (t < s) { r1[t] = fmaxf(r1[t], r1[t+s]); r2[t] = fmaxf(r2[t], r2[t+s]); }
    __syncthreads();
  }
  if (t == 0) { mx1[b] = r1[0]; mx2[b] = r2[0]; }
}

// ---------------- E1p/E1n factor arrays --------------------------------------
__global__ void k_exp(const float* __restrict__ s1, const float* __restrict__ mx1,
                      float* __restrict__ e1p, float* __restrict__ e1n) {
  int idx = blockIdx.x*256 + threadIdx.x;   // over B*N
  int b = idx / N_;
  float d = s1[idx] - mx1[b];               // <= 0
  e1p[idx] = expf(d);
  e1n[idx] = expf(0.01f*d);
}

// ---------------- row_unnorm[i] = Ai*sum_pos E1p + Bi*sum_neg E1n -------------
__global__ void k_row(const float* __restrict__ s1, const float* __restrict__ s2,
                      const float* __restrict__ e1p, const float* __restrict__ e1n,
                      const float* __restrict__ mx1, const float* __restrict__ mx2,
                      float* __restrict__ row_un, float* __restrict__ part) {
  __shared__ float shs[N_], shp[N_], shn[N_];     // 24 KB of 320 KB
  const int nb = N_/256;                          // 8 blocks per batch
  int b = blockIdx.x / nb;
  int i = (blockIdx.x % nb)*256 + threadIdx.x;
  for (int j = threadIdx.x; j < N_; j += 256) {
    shs[j] = s1[b*N_+j];
    shp[j] = e1p[b*N_+j];
    shn[j] = e1n[b*N_+j];
  }
  __syncthreads();
  float m1 = mx1[b], m2 = mx2[b];
  float X  = m1 + m2;
  float Mx = (X > 0.f) ? X : 0.01f*X;             // global max of lrelu(eij)
  float s2i = s2[b*N_+i];
  float Ai = expf(s2i + m1 - Mx);                 // exponent <= 0
  float Bi = expf(0.01f*(s2i + m1) - Mx);         // exponent <= 0
  float thr = -s2i;
  float ap = 0.f, an = 0.f;
  #pragma unroll 4
  for (int j = 0; j < N_; ++j) {
    bool pos = shs[j] > thr;                      // s2i + s1j > 0 ?
    ap += pos ? shp[j] : 0.f;
    an += pos ? 0.f : shn[j];
  }
  float r = Ai*ap + Bi*an;
  row_un[b*N_+i] = r;
  __shared__ float red[256];
  red[threadIdx.x] = r; __syncthreads();
  for (int s = 128; s > 0; s >>= 1) {
    if (threadIdx.x < s) red[threadIdx.x] += red[threadIdx.x+s];
    __syncthreads();
  }
  if (threadIdx.x == 0) part[blockIdx.x] = red[0];
}

// ---------------- deterministic denom reduction -------------------------------
__global__ void k_denom(const float* __restrict__ part, float* __restrict__ denom) {
  int b = threadIdx.x;
  if (b < B_) {
    float s = 0.f;
    for (int i = 0; i < N_/256; ++i) s += part[b*(N_/256)+i];
    denom[b] = s;
  }
}

// ---------------- h_t = z * (row/denom) * mask --------------------------------
__global__ void k_scale(const float* __restrict__ z, const float* __restrict__ row_un,
                        const float* __restrict__ denom, const float* __restrict__ mask,
                        float* __restrict__ ht) {
  size_t idx = (size_t)blockIdx.x*256 + threadIdx.x;  // B*N*H
  size_t node = idx >> 7;
  int b = (int)(node >> 11);
  ht[idx] = z[idx] * (row_un[node] / denom[b]) * mask[node];
}

// ---------------- cat = [h0 | h_t] -------------------------------------------
__global__ void k_concat(const float* __restrict__ h0, const float* __restrict__ ht,
                         float* __restrict__ cat) {
  size_t idx = (size_t)blockIdx.x*256 + threadIdx.x;  // M*256
  size_t m = idx >> 8;
  int c = (int)(idx & 255);
  cat[idx] = (c < H_) ? h0[m*H_ + c] : ht[m*H_ + (c - H_)];
}

// ---------------- out[b,l] = sum_n sigmoid(gate)*val*mask ---------------------
__global__ void k_final(const float* __restrict__ gate, const float* __restrict__ val,
                        const float* __restrict__ mask, float* __restrict__ out) {
  int b = blockIdx.x / LT_;
  int l = blockIdx.x % LT_;
  float acc = 0.f;
  for (int n = threadIdx.x; n < N_; n += 256) {
    size_t node = (size_t)b*N_ + n;
    float g = gate[node*LT_ + l];
    float sg = 1.f / (1.f + expf(-g));
    acc += sg * val[node*LT_ + l] * mask[node];
  }
  __shared__ float red[256];
  red[threadIdx.x] = acc; __syncthreads();
  for (int s = 128; s > 0; s >>= 1) {
    if (threadIdx.x < s) red[threadIdx.x] += red[threadIdx.x+s];
    __syncthreads();
  }
  if (threadIdx.x == 0) out[blockIdx.x] = red[0];
}

extern "C" void kernel_launch(void* const* d_in, const int* in_sizes, int n_in,
                              void* d_out, int out_size, void* d_ws, size_t ws_size,
                              hipStream_t stream) {
  (void)in_sizes; (void)n_in; (void)out_size; (void)ws_size;
  // dict order: g, h_in, e, fc_w, attn_w, i_params(8), j_params(8)
  const float* h_in = (const float*)d_in[1];
  const float* fc_w = (const float*)d_in[3];
  const float* attn = (const float*)d_in[4];
  const float* iW1=(const float*)d_in[5],  *ib1=(const float*)d_in[6];
  const float* iW2=(const float*)d_in[7],  *ib2=(const float*)d_in[8];
  const float* iW3=(const float*)d_in[9],  *ib3=(const float*)d_in[10];
  const float* iW4=(const float*)d_in[11], *ib4=(const float*)d_in[12];
  const float* jW1=(const float*)d_in[13], *jb1=(const float*)d_in[14];
  const float* jW2=(const float*)d_in[15], *jb2=(const float*)d_in[16];
  const float* jW3=(const float*)d_in[17], *jb3=(const float*)d_in[18];
  const float* jW4=(const float*)d_in[19], *jb4=(const float*)d_in[20];
  float* out = (float*)d_out;

  float* ws = (float*)d_ws;
  size_t off = 0;
  auto alloc = [&](size_t nelem) { float* p = ws + off; off += nelem; return p; };
  float* h0   = alloc((size_t)M_*H_);
  float* ht   = alloc((size_t)M_*H_);
  float* z    = alloc((size_t)M_*H_);
  float* cat  = alloc((size_t)M_*2*H_);
  float* tA   = alloc((size_t)M_*256);
  float* tB   = alloc((size_t)M_*256);
  float* gate = alloc((size_t)M_*LT_);
  float* val  = alloc((size_t)M_*LT_);
  float* mask = alloc(M_);
  float* s1   = alloc(M_);
  float* s2   = alloc(M_);
  float* e1p  = alloc(M_);
  float* e1n  = alloc(M_);
  float* row  = alloc(M_);
  float* mx1  = alloc(B_);
  float* mx2  = alloc(B_);
  float* den  = alloc(B_);
  float* part = alloc(64);

  auto gemm = [&](const float* X, const float* W, const float* bias, float* C,
                  int K, int NoutPad, int NoutValid, int relu) {
    if ((NoutPad & 63) == 0) {
      int waves = (M_/16) * (NoutPad/64);         // 16x64 strip per wave
      k_gemm_wmma4<<<dim3(waves/8), dim3(256), 0, stream>>>(X, W, bias, C, K, NoutPad, relu);
    } else {
      int waves = (M_/16);                        // 16x16 tile per wave (Nout<=16)
      k_gemm_wmma1<<<dim3(waves/8), dim3(256), 0, stream>>>(X, W, bias, C, K, NoutValid, relu);
    }
  };

  k_setup<<<M_, H_, 0, stream>>>(h_in, h0, ht, mask);

  for (int layer = 0; layer < 3; ++layer) {
    gemm(ht, fc_w, nullptr, z, H_, H_, H_, 0);
    k_attn_scores<<<B_, 256, 0, stream>>>(z, attn, s1, s2, mx1, mx2);
    k_exp<<<M_/256, 256, 0, stream>>>(s1, mx1, e1p, e1n);
    k_row<<<B_*(N_/256), 256, 0, stream>>>(s1, s2, e1p, e1n, mx1, mx2, row, part);
    k_denom<<<1, 32, 0, stream>>>(part, den);
    k_scale<<<(M_*H_)/256, 256, 0, stream>>>(z, row, den, mask, ht);
  }

  k_concat<<<(M_*2*H_)/256, 256, 0, stream>>>(h0, ht, cat);

  // gate = sigmoid(MLP_i([h0|ht]))  (sigmoid folded into k_final)
  gemm(cat, iW1, ib1, tA, 2*H_, 128, 128, 1);
  gemm(tA,  iW2, ib2, tB, 128, 256, 256, 1);
  gemm(tB,  iW3, ib3, tA, 256, 128, 128, 1);
  gemm(tA,  iW4, ib4, gate, 128, 16, LT_, 0);
  // val = MLP_j(ht)
  gemm(ht,  jW1, jb1, tA, 128, 128, 128, 1);
  gemm(tA,  jW2, jb2, tB, 128, 256, 256, 1);
  gemm(tB,  jW3, jb3, tA, 256, 128, 128, 1);
  gemm(tA,  jW4, jb4, val, 128, 16, LT_, 0);

  k_final<<<B_*LT_, 256, 0, stream>>>(gate, val, mask, out);
}